// GAT_7739531067833
// MI455X (gfx1250) — compile-verified
//
#include <hip/hip_runtime.h>
#include <hip/hip_bf16.h>
#include <math.h>

typedef __attribute__((ext_vector_type(2))) float v2f;
typedef __attribute__((ext_vector_type(8))) float v8f;

#define NEG_SLOPE 0.2f

// ---------- helpers ----------
__device__ __forceinline__ float lrelu(float v) { return v > 0.f ? v : NEG_SLOPE * v; }

// Order-preserving float -> uint encoding so atomicMax(u32) == float max.
__device__ __forceinline__ unsigned fenc(float f) {
  unsigned u = __float_as_uint(f);
  return (u & 0x80000000u) ? ~u : (u | 0x80000000u);
}
__device__ __forceinline__ float fdec(unsigned u) {
  return __uint_as_float((u & 0x80000000u) ? (u & 0x7FFFFFFFu) : ~u);
}

__device__ __forceinline__ void edge_nodes(const long long* __restrict__ ei, int e, int E,
                                           int& s, int& d) {
  if (e < E) { s = (int)ei[e]; d = (int)ei[(long long)E + e]; }
  else       { s = d = e - E; }                  // self-loops appended after E edges
}

// ---------- WMMA GEMM: C[M x Nc] = A[M x K] * Bp[K x NT*16], fp32 16x16x4 ----------
// Bp MUST be padded to exactly NT*16 columns (zero fill) so the inner loop has
// no lane-divergent guards: pure global_load + v_wmma, EXEC untouched.
// A-frag (ISA 7.12.2, 32-bit 16x4): lane<16 -> K={0,1}, lane>=16 -> K={2,3}, row = lane&15.
// B-frag mirrored; C/D: VGPR i -> row (i + 8*(lane>=16)), col = lane&15.
template <int NT>
__global__ void wmma_gemm(const float* __restrict__ A, const float* __restrict__ Bp,
                          float* __restrict__ C, int M, int K, int Nc) {
  const int BW = NT * 16;                        // padded B width
  int lane  = threadIdx.x & 31;
  int wave  = threadIdx.x >> 5;
  int mtile = blockIdx.x * (blockDim.x >> 5) + wave;
  if (mtile * 16 >= M) return;                   // wave-uniform exit: EXEC all-1s for WMMA

  int kk  = (lane >> 4) << 1;                    // 0 or 2
  int col = lane & 15;
  int row = mtile * 16 + col;

  v8f acc[NT] = {};
  const float* Arow = A + (long long)row * K + kk;     // a.x at [k], a.y at [k+1]
  const float* B0   = Bp + (long long)kk * BW + col;   // b.x at [k*BW], b.y at [(k+1)*BW]

#pragma unroll 4
  for (int k = 0; k < K; k += 4) {
    v2f a;
    a.x = Arow[k];
    a.y = Arow[k + 1];
    v2f b[NT];
#pragma unroll
    for (int t = 0; t < NT; ++t) {
      b[t].x = B0[(long long)k * BW + t * 16];
      b[t].y = B0[(long long)(k + 1) * BW + t * 16];
    }
#pragma unroll
    for (int t = 0; t < NT; ++t)
      acc[t] = __builtin_amdgcn_wmma_f32_16x16x4_f32(
          /*neg_a=*/false, a, /*neg_b=*/false, b[t],
          /*c_mod=*/(short)0, acc[t], /*reuse_a=*/false, /*reuse_b=*/false);
  }

  // store (runtime Nc guard only here, once per wave)
  int rbase = mtile * 16 + ((lane >> 4) << 3);
#pragma unroll
  for (int t = 0; t < NT; ++t) {
    int c = t * 16 + col;
    if (c < Nc) {
#pragma unroll
      for (int i = 0; i < 8; ++i)
        C[(long long)(rbase + i) * Nc + c] = acc[t][i];
    }
  }
}

// ---------- zero-pad a K x Nc matrix to K x BW ----------
__global__ void pad_mat(const float* __restrict__ W, float* __restrict__ Wp,
                        int K, int Nc, int BW) {
  int i = blockIdx.x * blockDim.x + threadIdx.x;
  if (i >= K * BW) return;
  int r = i / BW, c = i % BW;
  Wp[i] = (c < Nc) ? W[r * Nc + c] : 0.f;
}

// ---------- attention logits: al_s[n,h] = <h[n,h,:], a_src[h,:]>, same for dst ----------
__global__ void attn_logits(const float* __restrict__ h, const float* __restrict__ a_src,
                            const float* __restrict__ a_dst,
                            float* __restrict__ als, float* __restrict__ ald,
                            int N, int Hh, int F) {
  int idx = blockIdx.x * blockDim.x + threadIdx.x;
  if (idx >= N * Hh) return;
  int n = idx / Hh, hd = idx % Hh;
  const float* hp = h + ((long long)n * Hh + hd) * F;
  float s = 0.f, d = 0.f;
  for (int f = 0; f < F; ++f) {
    float v = hp[f];
    s += v * a_src[hd * F + f];
    d += v * a_dst[hd * F + f];
  }
  als[idx] = s;
  ald[idx] = d;
}

// ---------- edge pass 1: per-destination max of leaky_relu logits ----------
__global__ void edge_max(const long long* __restrict__ ei, int E, long long total, int Hh,
                         const float* __restrict__ als, const float* __restrict__ ald,
                         unsigned* __restrict__ menc) {
  long long idx = (long long)blockIdx.x * blockDim.x + threadIdx.x;
  if (idx >= total) return;
  int e = (int)(idx / Hh), hd = (int)(idx % Hh);
  int s, d; edge_nodes(ei, e, E, s, d);
  float v = lrelu(als[s * Hh + hd] + ald[d * Hh + hd]);
  atomicMax(&menc[d * Hh + hd], fenc(v));
}

// ---------- edge pass 2: per-destination sum of exp(e - max) ----------
__global__ void edge_expsum(const long long* __restrict__ ei, int E, long long total, int Hh,
                            const float* __restrict__ als, const float* __restrict__ ald,
                            const unsigned* __restrict__ menc, float* __restrict__ ssum) {
  long long idx = (long long)blockIdx.x * blockDim.x + threadIdx.x;
  if (idx >= total) return;
  int e = (int)(idx / Hh), hd = (int)(idx % Hh);
  int s, d; edge_nodes(ei, e, E, s, d);
  float v = lrelu(als[s * Hh + hd] + ald[d * Hh + hd]);
  float m = fdec(menc[d * Hh + hd]);
  atomicAdd(&ssum[d * Hh + hd], expf(v - m));
}

// ---------- edge pass 3: agg[dst,h,f] += h[src,h,f] * alpha ----------
__global__ void edge_agg(const long long* __restrict__ ei, int E, long long total, int Hh, int F,
                         const float* __restrict__ als, const float* __restrict__ ald,
                         const unsigned* __restrict__ menc, const float* __restrict__ ssum,
                         const float* __restrict__ h, float* __restrict__ agg) {
  long long idx = (long long)blockIdx.x * blockDim.x + threadIdx.x;
  if (idx >= total) return;
  int HF = Hh * F;
  int e  = (int)(idx / HF);
  int r  = (int)(idx % HF);
  int hd = r / F, f = r % F;
  int s, d; edge_nodes(ei, e, E, s, d);
  float v = lrelu(als[s * Hh + hd] + ald[d * Hh + hd]);
  float m = fdec(menc[d * Hh + hd]);
  float alpha = expf(v - m) / (ssum[d * Hh + hd] + 1e-16f);
  atomicAdd(&agg[((long long)d * Hh + hd) * F + f],
            h[((long long)s * Hh + hd) * F + f] * alpha);
}

// ---------- bias + ELU ----------
__global__ void bias_elu(const float* __restrict__ agg, const float* __restrict__ bias,
                         float* __restrict__ out, int total, int C) {
  int i = blockIdx.x * blockDim.x + threadIdx.x;
  if (i >= total) return;
  float v = agg[i] + bias[i % C];
  out[i] = v > 0.f ? v : (expf(v) - 1.f);
}

// ---------- bias + log_softmax (one thread per node, C=40) ----------
__global__ void bias_logsoftmax(const float* __restrict__ acc, const float* __restrict__ bias,
                                float* __restrict__ out, int N, int C) {
  int n = blockIdx.x * blockDim.x + threadIdx.x;
  if (n >= N) return;
  const float* row = acc + (long long)n * C;
  float mx = -INFINITY;
  for (int c = 0; c < C; ++c) mx = fmaxf(mx, row[c] + bias[c]);
  float s = 0.f;
  for (int c = 0; c < C; ++c) s += expf(row[c] + bias[c] - mx);
  float ls = logf(s);
  for (int c = 0; c < C; ++c) out[(long long)n * C + c] = row[c] + bias[c] - mx - ls;
}

// ---------- launcher ----------
extern "C" void kernel_launch(void* const* d_in, const int* in_sizes, int n_in,
                              void* d_out, int out_size, void* d_ws, size_t ws_size,
                              hipStream_t stream) {
  const float*     x     = (const float*)d_in[0];
  const long long* ei    = (const long long*)d_in[1];   // int64 edge_index [2, E]
  const float*     W1    = (const float*)d_in[2];
  const float*     asrc1 = (const float*)d_in[3];
  const float*     adst1 = (const float*)d_in[4];
  const float*     b1    = (const float*)d_in[5];
  const float*     W2    = (const float*)d_in[6];
  const float*     asrc2 = (const float*)d_in[7];
  const float*     adst2 = (const float*)d_in[8];
  const float*     b2    = (const float*)d_in[9];
  float*           out   = (float*)d_out;

  const int IN = 512, H = 8, F1 = 8, C1 = H * F1, OC = 40, OCP = 48;
  const int N    = in_sizes[0] / IN;
  const int E    = in_sizes[1] / 2;
  const int Etot = E + N;                     // with self-loops

  // workspace layout (all fp32-sized slots, N multiples => 16B aligned)
  float* ws = (float*)d_ws;
  float*    h1    = ws;              ws += (size_t)N * C1;
  float*    als1  = ws;              ws += (size_t)N * H;
  float*    ald1  = ws;              ws += (size_t)N * H;
  unsigned* menc1 = (unsigned*)ws;   ws += (size_t)N * H;
  float*    ssum1 = ws;              ws += (size_t)N * H;
  float*    agg1  = ws;              ws += (size_t)N * C1;
  float*    h1a   = ws;              ws += (size_t)N * C1;
  float*    h2    = ws;              ws += (size_t)N * OC;
  float*    als2  = ws;              ws += (size_t)N;
  float*    ald2  = ws;              ws += (size_t)N;
  unsigned* menc2 = (unsigned*)ws;   ws += (size_t)N;
  float*    ssum2 = ws;              ws += (size_t)N;
  float*    agg2  = ws;              ws += (size_t)N * OC;
  float*    W2p   = ws;              ws += (size_t)C1 * OCP;   // W2 zero-padded 64 x 48

  hipMemsetAsync(menc1, 0, (size_t)N * H  * 4, stream);   // fenc inverse of 0 == -inf
  hipMemsetAsync(ssum1, 0, (size_t)N * H  * 4, stream);
  hipMemsetAsync(agg1,  0, (size_t)N * C1 * 4, stream);
  hipMemsetAsync(menc2, 0, (size_t)N      * 4, stream);
  hipMemsetAsync(ssum2, 0, (size_t)N      * 4, stream);
  hipMemsetAsync(agg2,  0, (size_t)N * OC * 4, stream);

  const int mtiles = (N + 15) / 16;           // 50000/16 = 3125 exact
  const int WPB    = 8;                       // 8 waves (wave32) per block
  dim3 gblk(32 * WPB);
  dim3 ggrd((mtiles + WPB - 1) / WPB);

  // ===== layer 1: 512 -> 8 heads x 8 (W1 is 512x64 == 4 full tiles, no padding) =====
  wmma_gemm<4><<<ggrd, gblk, 0, stream>>>(x, W1, h1, N, IN, C1);
  {
    int t = N * H;
    attn_logits<<<(t + 255) / 256, 256, 0, stream>>>(h1, asrc1, adst1, als1, ald1, N, H, F1);
  }
  {
    long long t = (long long)Etot * H;
    int g = (int)((t + 255) / 256);
    edge_max   <<<g, 256, 0, stream>>>(ei, E, t, H, als1, ald1, menc1);
    edge_expsum<<<g, 256, 0, stream>>>(ei, E, t, H, als1, ald1, menc1, ssum1);
  }
  {
    long long t = (long long)Etot * H * F1;
    edge_agg<<<(int)((t + 255) / 256), 256, 0, stream>>>(ei, E, t, H, F1,
                                                         als1, ald1, menc1, ssum1, h1, agg1);
  }
  {
    int t = N * C1;
    bias_elu<<<(t + 255) / 256, 256, 0, stream>>>(agg1, b1, h1a, t, C1);
  }

  // ===== layer 2: 64 -> 1 head x 40 (pad W2 to 64x48 for unguarded B loads) =====
  {
    int t = C1 * OCP;
    pad_mat<<<(t + 255) / 256, 256, 0, stream>>>(W2, W2p, C1, OC, OCP);
  }
  wmma_gemm<3><<<ggrd, gblk, 0, stream>>>(h1a, W2p, h2, N, C1, OC);
  attn_logits<<<(N + 255) / 256, 256, 0, stream>>>(h2, asrc2, adst2, als2, ald2, N, 1, OC);
  {
    long long t = (long long)Etot;
    int g = (int)((t + 255) / 256);
    edge_max   <<<g, 256, 0, stream>>>(ei, E, t, 1, als2, ald2, menc2);
    edge_expsum<<<g, 256, 0, stream>>>(ei, E, t, 1, als2, ald2, menc2, ssum2);
  }
  {
    long long t = (long long)Etot * OC;
    edge_agg<<<(int)((t + 255) / 256), 256, 0, stream>>>(ei, E, t, 1, OC,
                                                         als2, ald2, menc2, ssum2, h2, agg2);
  }
  bias_logsoftmax<<<(N + 127) / 128, 128, 0, stream>>>(agg2, b2, out, N, OC);
}